// GraphConvLayer_83416854823498
// MI455X (gfx1250) — compile-verified
//
#include <hip/hip_runtime.h>

typedef __attribute__((ext_vector_type(2))) float v2f;
typedef __attribute__((ext_vector_type(8))) float v8f;

#define TT     4
#define NNODE  50000
#define DH     64
#define EDGES  800000
#define MROWS  (TT * NNODE)        // 200000
#define KDIM   192
#define BN_EPS 1e-5f

// ---- workspace layout (float offsets into d_ws) ----
#define Z_OFF     0                               // NNODE floats
#define AGG_OFF   50048                           // TT*NNODE*DH floats (aligned)
#define AGG_CNT   (TT * NNODE * DH)               // 12,800,000
#define SUM_OFF   (AGG_OFF + AGG_CNT)             // 192
#define SUMSQ_OFF (SUM_OFF + KDIM)                // 192
#define WSC_OFF   (SUMSQ_OFF + KDIM)              // 192*64 scaled weights [k][n]
#define B2_OFF    (WSC_OFF + KDIM * DH)           // 64 fused bias
#define ZERO_CNT  (SUMSQ_OFF + KDIM)              // everything accumulated via atomics

// ------------------------------------------------------------------
__global__ void zero_kernel(float* ws, long count) {
    long i = (long)blockIdx.x * blockDim.x + threadIdx.x;
    long stride = (long)gridDim.x * blockDim.x;
    for (; i < count; i += stride) ws[i] = 0.0f;
}

// Z[dst] += ew
__global__ void zacc_kernel(const float* __restrict__ ew, const int* __restrict__ dst,
                            float* __restrict__ Z) {
    int e = blockIdx.x * 256 + threadIdx.x;
    if (e < EDGES) atomicAdd(&Z[dst[e]], ew[e]);
}

// agg[t, dst[e], c] += ew[e] * node[t, src[e], c]
__global__ __launch_bounds__(256) void scatter_kernel(const float* __restrict__ node,
                                                      const float* __restrict__ ew,
                                                      const int* __restrict__ src,
                                                      const int* __restrict__ dst,
                                                      float* __restrict__ agg) {
    int e = blockIdx.x * 4 + (threadIdx.x >> 6);
    int c = threadIdx.x & 63;
    int t = blockIdx.y;
    if (e < EDGES) {
        float w = ew[e];
        int s = src[e], d = dst[e];
        float v = node[((long)t * NNODE + s) * DH + c] * w;
        atomicAdd(&agg[((long)t * NNODE + d) * DH + c], v);
    }
}

// agg /= max(Z,0->1), in place
__global__ void normagg_kernel(float* __restrict__ agg, const float* __restrict__ Z) {
    long i = (long)blockIdx.x * 256 + threadIdx.x;
    if (i < (long)AGG_CNT) {
        int n = (int)((i >> 6) % NNODE);
        float z = Z[n];
        z = (z == 0.0f) ? 1.0f : z;
        agg[i] /= z;
    }
}

// per-channel sum / sumsq of the implicit [MROWS, 192] input
__global__ __launch_bounds__(192) void colstats_kernel(const float* __restrict__ node,
                                                       const float* __restrict__ agg,
                                                       float* __restrict__ ws) {
    int c = threadIdx.x;          // 0..191
    int seg = c >> 6;
    int off = c & 63;
    float s = 0.0f, sq = 0.0f;
    for (long row = blockIdx.x; row < MROWS; row += gridDim.x) {
        int t = (int)(row / NNODE);
        int n = (int)(row - (long)t * NNODE);
        float v;
        if (seg == 0)      v = node[((long)t * NNODE + n) * DH + off];
        else if (seg == 1) v = (t > 0) ? node[((long)(t - 1) * NNODE + n) * DH + off] : 0.0f;
        else               v = agg[((long)t * NNODE + n) * DH + off];
        s += v;
        sq += v * v;
    }
    atomicAdd(&ws[SUM_OFF + c], s);
    atomicAdd(&ws[SUMSQ_OFF + c], sq);
}

// fold BN into weights: Wsc[c][k] = W1[k][c]*a[c], b2[k] = b1[k] + sum_c (beta-mu*a)[c]*W1[k][c]
__global__ __launch_bounds__(192) void finalize_kernel(const float* __restrict__ W1,
                                                       const float* __restrict__ b1,
                                                       const float* __restrict__ gamma,
                                                       const float* __restrict__ beta,
                                                       float* __restrict__ ws) {
    __shared__ float a_sh[KDIM];
    __shared__ float sh_sh[KDIM];
    int c = threadIdx.x;
    float inv_m = 1.0f / (float)MROWS;
    float mu = ws[SUM_OFF + c] * inv_m;
    float var = ws[SUMSQ_OFF + c] * inv_m - mu * mu;
    float sc = rsqrtf(var + BN_EPS) * gamma[c];
    a_sh[c] = sc;
    sh_sh[c] = beta[c] - mu * sc;
    __syncthreads();
    for (int k = 0; k < DH; ++k)
        ws[WSC_OFF + c * DH + k] = W1[(long)k * KDIM + c] * a_sh[c];
    if (c < DH) {
        float acc = b1[c];
        for (int j = 0; j < KDIM; ++j)
            acc += sh_sh[j] * W1[(long)c * KDIM + j];
        ws[B2_OFF + c] = acc;
    }
}

// ------------------------------------------------------------------
// GEMM: out[row, k] = relu( sum_c inp[row,c] * Wsc[c,k] + b2[k] )
// fp32 WMMA 16x16x4; each wave: 16 rows x 64 cols, K = 192 in 48 steps.
// B held in LDS as paired float2 {B[k][n], B[k+1][n]}.
__global__ __launch_bounds__(256) void gemm_kernel(const float* __restrict__ node,
                                                   const float* __restrict__ agg,
                                                   const float* __restrict__ Wsc,
                                                   const float* __restrict__ bias2,
                                                   float* __restrict__ out) {
    __shared__ float Bs[KDIM * DH];     // 48 KB, paired layout
    for (int i = threadIdx.x; i < KDIM * DH; i += 256) {
        int k = i >> 6, n = i & 63;
        Bs[(k >> 1) * 128 + n * 2 + (k & 1)] = Wsc[i];
    }
    __syncthreads();

    const int wave = threadIdx.x >> 5;
    const int lane = threadIdx.x & 31;
    const int m   = lane & 15;
    const int hi  = lane >> 4;
    const int hi2 = hi * 2;
    const long rowBase = (long)blockIdx.x * 128 + wave * 16;

    long row  = rowBase + m;
    long rowc = (row < MROWS) ? row : (long)(MROWS - 1);   // clamp: garbage rows never stored
    int t = (int)(rowc / NNODE);
    int n = (int)(rowc - (long)t * NNODE);
    const float* p_cur  = node + ((long)t * NNODE + n) * DH;
    const int    tp     = (t > 0) ? (t - 1) : 0;
    const float* p_prev = node + ((long)tp * NNODE + n) * DH;
    const float  prevFlag = (t > 0) ? 1.0f : 0.0f;
    const float* p_agg  = agg + ((long)t * NNODE + n) * DH;

    v8f a0 = {}, a1 = {}, a2 = {}, a3 = {};
    const v2f* Bs2 = (const v2f*)Bs;

#define WMMA_STEP(AV, BASEK, KK)                                                            \
    {                                                                                       \
        const v2f* bp = Bs2 + (((BASEK) + (KK) * 4 + hi2) >> 1) * 64;                       \
        v2f b0 = bp[m];                                                                     \
        v2f b1 = bp[16 + m];                                                                \
        v2f b2 = bp[32 + m];                                                                \
        v2f b3 = bp[48 + m];                                                                \
        a0 = __builtin_amdgcn_wmma_f32_16x16x4_f32(false, (AV), false, b0, (short)0, a0, false, false); \
        a1 = __builtin_amdgcn_wmma_f32_16x16x4_f32(false, (AV), false, b1, (short)0, a1, false, false); \
        a2 = __builtin_amdgcn_wmma_f32_16x16x4_f32(false, (AV), false, b2, (short)0, a2, false, false); \
        a3 = __builtin_amdgcn_wmma_f32_16x16x4_f32(false, (AV), false, b3, (short)0, a3, false, false); \
    }

#pragma unroll
    for (int kk = 0; kk < 16; ++kk) {                 // channels 0..63: node_data
        v2f av = *(const v2f*)(p_cur + kk * 4 + hi2);
        WMMA_STEP(av, 0, kk);
    }
#pragma unroll
    for (int kk = 0; kk < 16; ++kk) {                 // channels 64..127: prev (0 at t==0)
        v2f av = *(const v2f*)(p_prev + kk * 4 + hi2);
        av.x *= prevFlag;
        av.y *= prevFlag;
        WMMA_STEP(av, 64, kk);
    }
#pragma unroll
    for (int kk = 0; kk < 16; ++kk) {                 // channels 128..191: avg_neighbor
        v2f av = *(const v2f*)(p_agg + kk * 4 + hi2);
        WMMA_STEP(av, 128, kk);
    }
#undef WMMA_STEP

    float bb0 = bias2[m];
    float bb1 = bias2[16 + m];
    float bb2 = bias2[32 + m];
    float bb3 = bias2[48 + m];
#pragma unroll
    for (int i = 0; i < 8; ++i) {
        long r = rowBase + 8 * hi + i;
        if (r < MROWS) {
            float* o = out + r * DH + m;
            float v0 = a0[i] + bb0; o[0]  = v0 > 0.0f ? v0 : 0.0f;
            float v1 = a1[i] + bb1; o[16] = v1 > 0.0f ? v1 : 0.0f;
            float v2 = a2[i] + bb2; o[32] = v2 > 0.0f ? v2 : 0.0f;
            float v3 = a3[i] + bb3; o[48] = v3 > 0.0f ? v3 : 0.0f;
        }
    }
}

// ------------------------------------------------------------------
extern "C" void kernel_launch(void* const* d_in, const int* in_sizes, int n_in,
                              void* d_out, int out_size, void* d_ws, size_t ws_size,
                              hipStream_t stream) {
    const float* node  = (const float*)d_in[0];
    const float* ew    = (const float*)d_in[1];
    const float* W1    = (const float*)d_in[2];
    const float* b1    = (const float*)d_in[3];
    const float* gamma = (const float*)d_in[4];
    const float* beta  = (const float*)d_in[5];
    const int*   src   = (const int*)d_in[6];
    const int*   dst   = (const int*)d_in[7];
    float* ws = (float*)d_ws;
    float* out = (float*)d_out;

    float* Z     = ws + Z_OFF;
    float* agg   = ws + AGG_OFF;
    float* Wsc   = ws + WSC_OFF;
    float* bias2 = ws + B2_OFF;

    // 1. zero Z, agg, stats
    zero_kernel<<<2048, 256, 0, stream>>>(ws, (long)ZERO_CNT);
    // 2. degree sums
    zacc_kernel<<<(EDGES + 255) / 256, 256, 0, stream>>>(ew, dst, Z);
    // 3. weighted neighbor scatter
    scatter_kernel<<<dim3((EDGES + 3) / 4, TT), 256, 0, stream>>>(node, ew, src, dst, agg);
    // 4. normalize agg by degree
    normagg_kernel<<<(AGG_CNT + 255) / 256, 256, 0, stream>>>(agg, Z);
    // 5. batchnorm column statistics
    colstats_kernel<<<512, 192, 0, stream>>>(node, agg, ws);
    // 6. fold BN into weights + fused bias
    finalize_kernel<<<1, 192, 0, stream>>>(W1, b1, gamma, beta, ws);
    // 7. fused BN+GEMM+bias+ReLU via fp32 WMMA
    gemm_kernel<<<(MROWS + 127) / 128, 256, 0, stream>>>(node, agg, Wsc, bias2, out);
}